// MultiHeadAttentionOracle_70841190580863
// MI455X (gfx1250) — compile-verified
//
#include <hip/hip_runtime.h>
#include <hip/hip_bf16.h>

// ---------------- problem constants ----------------
#define WIDTH 64
#define SEQ   (WIDTH * WIDTH)   // 4096
#define EMB   128
#define BATCH 8
#define NEDGE 512               // 32 paths * 16 edges per batch
#define ROWS  (BATCH * SEQ)     // 32768

typedef __attribute__((ext_vector_type(2))) float v2f;
typedef __attribute__((ext_vector_type(8))) float v8f;

// ---------------------------------------------------------------------------
// Zero-fill (float4 vectorized, grid-stride) — capture-safe replacement for
// memset; dominates nothing since the scatter/GEMMs overwrite afterwards.
// ---------------------------------------------------------------------------
__global__ void fill_zero4(float4* __restrict__ p, size_t n4) {
    size_t i = blockIdx.x * (size_t)blockDim.x + threadIdx.x;
    size_t stride = (size_t)gridDim.x * blockDim.x;
    float4 z = make_float4(0.f, 0.f, 0.f, 0.f);
    for (; i < n4; i += stride) p[i] = z;
}

// ---------------------------------------------------------------------------
// Edge metadata: src/dst node ids + flags.
//   flag bit0 = duplicate (an earlier edge in this batch has same (src,dst))
//   flag bit1 = first edge in this batch with this src (the "owner")
// Deterministic O(E^2) scan per batch: 512*512 int ops — negligible.
// ---------------------------------------------------------------------------
__global__ void edge_meta(const int* __restrict__ edges,
                          int* __restrict__ src, int* __restrict__ dst,
                          int* __restrict__ flags) {
    int j = blockIdx.x * blockDim.x + threadIdx.x;
    if (j >= BATCH * NEDGE) return;
    int b = j >> 9, e = j & (NEDGE - 1);
    const int* p = edges + (size_t)j * 4;  // [x0,y0,x1,y1]
    int s = p[1] * WIDTH + p[0];
    int d = p[3] * WIDTH + p[2];
    src[j] = s;
    dst[j] = d;
    int base = b << 9;
    bool dup = false, first = true;
    for (int i = 0; i < e; ++i) {
        const int* q = edges + (size_t)(base + i) * 4;
        int si = q[1] * WIDTH + q[0];
        if (si == s) {
            first = false;
            int di = q[3] * WIDTH + q[2];
            if (di == d) dup = true;
        }
    }
    flags[j] = (dup ? 1 : 0) | (first ? 2 : 0);
}

// ---------------------------------------------------------------------------
// Scatter 1.0 into both adjacency copies (idempotent, matches .set(1.0)).
// ---------------------------------------------------------------------------
__global__ void scatter_A(const int* __restrict__ src, const int* __restrict__ dst,
                          float* __restrict__ A1, float* __restrict__ A2) {
    int j = blockIdx.x * blockDim.x + threadIdx.x;
    if (j >= BATCH * NEDGE) return;
    int b = j >> 9;
    size_t idx = ((size_t)b * SEQ + src[j]) * SEQ + dst[j];
    A1[idx] = 1.0f;
    A2[idx] = 1.0f;
}

// ---------------------------------------------------------------------------
// Dense GEMM  Y[R,128] = X[R,128] @ W[128,128] + bias[128]
// using V_WMMA_F32_16X16X4_F32 (wave32, 16x16 tile per wave, 8 N-tiles).
// Lane maps per CDNA5 ISA 7.12.2:
//   A 16x4 : lanes 0-15 -> M=lane, K=k+{0,1}; lanes 16-31 -> K=k+{2,3}
//   B 4x16 : lanes 0-15 -> N=lane, K=k+{0,1}; lanes 16-31 -> K=k+{2,3}
//   C 16x16: vgpr r, lanes 0-15 -> M=r; lanes 16-31 -> M=r+8
// blockDim = 256 (8 waves); each wave owns a 16-row strip; grid = R/128.
// ---------------------------------------------------------------------------
__global__ __launch_bounds__(256) void gemm128_wmma(
    const float* __restrict__ X, const float* __restrict__ W,
    const float* __restrict__ bias, float* __restrict__ Y) {
    const int E = EMB;
    int wave = threadIdx.x >> 5;
    int lane = threadIdx.x & 31;
    int lo = lane & 15;
    int hi = lane >> 4;
    size_t m0 = ((size_t)blockIdx.x * 8 + wave) * 16;
    const float* xrow = X + (m0 + lo) * E;
    int koff = hi * 2;

    v8f acc[8];
#pragma unroll
    for (int n = 0; n < 8; ++n) {
        float bval = bias[n * 16 + lo];   // C(M,N) = bias[N]
#pragma unroll
        for (int r = 0; r < 8; ++r) acc[n][r] = bval;
    }

#pragma unroll 4
    for (int k = 0; k < E; k += 4) {
        v2f a = *(const v2f*)(xrow + k + koff);        // aligned b64 load
        const float* wrow = W + (size_t)(k + koff) * E;
#pragma unroll
        for (int n = 0; n < 8; ++n) {
            v2f bfrag;
            bfrag[0] = wrow[n * 16 + lo];
            bfrag[1] = wrow[E + n * 16 + lo];
            acc[n] = __builtin_amdgcn_wmma_f32_16x16x4_f32(
                false, a, false, bfrag, (short)0, acc[n], false, false);
        }
    }

#pragma unroll
    for (int n = 0; n < 8; ++n) {
#pragma unroll
        for (int r = 0; r < 8; ++r) {
            Y[(m0 + r + 8 * hi) * E + n * 16 + lo] = acc[n][r];
        }
    }
}

// ---------------------------------------------------------------------------
// Sparse hidden = A @ v, deterministic (no atomics).
// One 128-thread block per edge; only the "owner" edge (first of its src)
// proceeds: it sums v[dst] over all distinct (src,dst) edges with this src,
// in fixed edge order, and plain-stores the row.
// ---------------------------------------------------------------------------
__global__ __launch_bounds__(128) void hidden_accum(
    const int* __restrict__ src, const int* __restrict__ dst,
    const int* __restrict__ flags, const float* __restrict__ v,
    float* __restrict__ hidden) {
    int j = blockIdx.x;            // 0 .. BATCH*NEDGE-1
    int b = j >> 9;
    int e0 = j & (NEDGE - 1);
    int base = b << 9;

    __shared__ int ssrc[NEDGE], sdst[NEDGE], sfl[NEDGE];
    for (int i = threadIdx.x; i < NEDGE; i += blockDim.x) {
        ssrc[i] = src[base + i];
        sdst[i] = dst[base + i];
        sfl[i]  = flags[base + i];
    }
    __syncthreads();

    if (!(sfl[e0] & 2)) return;    // not the owner of this src row
    int s = ssrc[e0];
    int e = threadIdx.x;           // embedding element 0..127
    float sum = 0.f;
    for (int i = e0; i < NEDGE; ++i) {
        if (ssrc[i] == s && !(sfl[i] & 1)) {
            sum += v[((size_t)b * SEQ + sdst[i]) * EMB + e];
        }
    }
    hidden[((size_t)b * SEQ + s) * EMB + e] = sum;
}

// ---------------------------------------------------------------------------
extern "C" void kernel_launch(void* const* d_in, const int* in_sizes, int n_in,
                              void* d_out, int out_size, void* d_ws, size_t ws_size,
                              hipStream_t stream) {
    // setup_inputs order:
    // 0 queries, 1 keys, 2 values, 3 oracle_edges, 4 Wq, 5 bq, 6 Wk, 7 bk,
    // 8 Wv, 9 bv, 10 Wp, 11 bp
    const float* values = (const float*)d_in[2];
    const int*   edges  = (const int*)d_in[3];
    const float* Wv     = (const float*)d_in[8];
    const float* bv     = (const float*)d_in[9];
    const float* Wp     = (const float*)d_in[10];
    const float* bp     = (const float*)d_in[11];

    float* out  = (float*)d_out;
    float* attn = out;                                   // [B,S,E]
    float* A1   = out + (size_t)ROWS * EMB;              // [B,S,S]
    float* A2   = A1 + (size_t)BATCH * SEQ * SEQ;        // [B,S,S]

    // workspace layout
    float* v_ws    = (float*)d_ws;                       // 16 MB
    float* hid_ws  = v_ws + (size_t)ROWS * EMB;          // 16 MB
    int*   meta    = (int*)(hid_ws + (size_t)ROWS * EMB);
    int*   src     = meta;                               // [B*NEDGE]
    int*   dst     = meta + BATCH * NEDGE;
    int*   flg     = meta + 2 * BATCH * NEDGE;

    const size_t aElems4 = (2ull * BATCH * SEQ * SEQ) / 4;  // both A copies
    const size_t hElems4 = ((size_t)ROWS * EMB) / 4;

    // 1) zero adjacency (1 GB) and hidden buffer
    hipLaunchKernelGGL(fill_zero4, dim3(8192), dim3(256), 0, stream,
                       (float4*)A1, aElems4);
    hipLaunchKernelGGL(fill_zero4, dim3(256), dim3(256), 0, stream,
                       (float4*)hid_ws, hElems4);

    // 2) edge metadata (dedup + owner flags), then scatter ones into A1/A2
    hipLaunchKernelGGL(edge_meta, dim3((BATCH * NEDGE + 255) / 256), dim3(256),
                       0, stream, edges, src, dst, flg);
    hipLaunchKernelGGL(scatter_A, dim3((BATCH * NEDGE + 255) / 256), dim3(256),
                       0, stream, src, dst, A1, A2);

    // 3) v = values @ Wv + bv   (WMMA f32)
    hipLaunchKernelGGL(gemm128_wmma, dim3(ROWS / 128), dim3(256), 0, stream,
                       values, Wv, bv, v_ws);

    // 4) hidden = A @ v  (sparse, deterministic)
    hipLaunchKernelGGL(hidden_accum, dim3(BATCH * NEDGE), dim3(128), 0, stream,
                       src, dst, flg, v_ws, hid_ws);

    // 5) attn = hidden @ Wp + bp  (WMMA f32)
    hipLaunchKernelGGL(gemm128_wmma, dim3(ROWS / 128), dim3(256), 0, stream,
                       hid_ws, Wp, bp, attn);
}